// GN_55714315764196
// MI455X (gfx1250) — compile-verified
//
#include <hip/hip_runtime.h>

#define D 128
#define NN 50000
#define NE 625000

typedef __attribute__((ext_vector_type(2))) float v2f;
typedef __attribute__((ext_vector_type(8))) float v8f;

// Hardware f32 atomic-add, guaranteed (no CAS-loop fallback). Non-returning
// form -> tracked by STOREcnt; s_endpgm's implicit wait covers completion.
__device__ __forceinline__ void atomic_add_f32_hw(float* addr, float val) {
  asm volatile("global_atomic_add_f32 %0, %1, off"
               :
               : "v"(addr), "v"(val)
               : "memory");
}

// ---------------------------------------------------------------------------
// Kernel 1: zero the workspace (summed[N,D] + deg[N]) each call (harness
// poisons ws once and never re-poisons; we must re-zero deterministically).
// ---------------------------------------------------------------------------
__global__ void GN_zero_ws(float* __restrict__ p, long n) {
  long i = (long)blockIdx.x * blockDim.x + threadIdx.x;
  long stride = (long)gridDim.x * blockDim.x;
  for (; i < n; i += stride) p[i] = 0.0f;
}

// ---------------------------------------------------------------------------
// Kernel 2: edge scatter. Grid-stride loop, one wave32 per edge per step.
// Lane l handles row elements l, l+32, l+64, l+96 -> coalesced 128B loads
// from x[src] and coalesced hardware f32 atomics into summed[dst].
// Lane 0 bumps the degree counter.
// ---------------------------------------------------------------------------
__global__ void GN_scatter(const float* __restrict__ x,
                           const int* __restrict__ src,
                           const int* __restrict__ dst,
                           float* __restrict__ summed,
                           float* __restrict__ deg) {
  int lane = threadIdx.x & 31;
  int wave = threadIdx.x >> 5;
  int wpb  = blockDim.x >> 5;
  long nwaves = (long)gridDim.x * wpb;

  for (long e = (long)blockIdx.x * wpb + wave; e < NE; e += nwaves) {
    int s = src[e];
    int d = dst[e];
    const float* __restrict__ xs = x + (long)s * D;
    float* __restrict__ sd = summed + (long)d * D;

#pragma unroll
    for (int p = 0; p < 4; ++p) {
      float v = xs[p * 32 + lane];
      atomic_add_f32_hw(&sd[p * 32 + lane], v);
    }
    if (lane == 0) {
      atomic_add_f32_hw(&deg[d], 1.0f);
    }
  }
}

// ---------------------------------------------------------------------------
// Kernel 3: fused epilogue GEMM via V_WMMA_F32_16X16X4_F32.
//   out = x @ W_self + (summed * 1/max(deg,1)) @ W_neigh + b
// One wave per 16x16 output tile; 8 waves/block cover 16 rows x 128 cols.
// A fragment (16x4 f32, 2 VGPRs/lane): lanes 0-15 = rows M=0..15 with K,K+1;
// lanes 16-31 = same rows with K+2,K+3. B fragment mirrors with N=lane%16.
// C/D: VGPR i = row m0+i (+8 for upper half-wave), col = n0 + lane%16.
// ---------------------------------------------------------------------------
__global__ void GN_sage_gemm(const float* __restrict__ x,
                             const float* __restrict__ summed,
                             const float* __restrict__ deg,
                             const float* __restrict__ Wself,
                             const float* __restrict__ Wneigh,
                             const float* __restrict__ bias,
                             float* __restrict__ out) {
  int wave = threadIdx.x >> 5;   // 0..7 -> column tile
  int lane = threadIdx.x & 31;
  int half = lane >> 4;          // 0: K,K+1   1: K+2,K+3
  int l16  = lane & 15;

  int m0 = blockIdx.x * 16;      // 50000 / 16 = 3125 exact, no tail
  int n0 = wave * 16;

  int row = m0 + l16;            // A-matrix row owned by this lane
  int col = n0 + l16;            // B/C/D column owned by this lane

  // deg is an integer-valued count; v_rcp_f32 (~1 ulp) is plenty for a mean.
  float invd = __builtin_amdgcn_rcpf(fmaxf(deg[row], 1.0f));

  const float* __restrict__ xr = x      + (long)row * D;
  const float* __restrict__ sr = summed + (long)row * D;

  v8f c = {};

  // self term: x @ W_self
#pragma unroll 4
  for (int k = 0; k < D; k += 4) {
    int kk = k + half * 2;
    v2f a, b;
    a.x = xr[kk];
    a.y = xr[kk + 1];
    b.x = Wself[(long)kk * D + col];
    b.y = Wself[(long)(kk + 1) * D + col];
    c = __builtin_amdgcn_wmma_f32_16x16x4_f32(false, a, false, b,
                                              (short)0, c, false, false);
  }

  // neighbor term: (summed * invd) @ W_neigh
#pragma unroll 4
  for (int k = 0; k < D; k += 4) {
    int kk = k + half * 2;
    v2f a, b;
    a.x = sr[kk] * invd;
    a.y = sr[kk + 1] * invd;
    b.x = Wneigh[(long)kk * D + col];
    b.y = Wneigh[(long)(kk + 1) * D + col];
    c = __builtin_amdgcn_wmma_f32_16x16x4_f32(false, a, false, b,
                                              (short)0, c, false, false);
  }

  float bv = bias[col];
#pragma unroll
  for (int i = 0; i < 8; ++i) {
    int m = m0 + i + half * 8;
    out[(long)m * D + col] = c[i] + bv;
  }
}

// ---------------------------------------------------------------------------
extern "C" void kernel_launch(void* const* d_in, const int* in_sizes, int n_in,
                              void* d_out, int out_size, void* d_ws, size_t ws_size,
                              hipStream_t stream) {
  const float* x      = (const float*)d_in[0];
  const int*   src    = (const int*)  d_in[1];
  const int*   dst    = (const int*)  d_in[2];
  const float* Wself  = (const float*)d_in[3];
  const float* Wneigh = (const float*)d_in[4];
  const float* bias   = (const float*)d_in[5];
  float*       out    = (float*)d_out;

  float* summed = (float*)d_ws;                 // N*D floats = 25.6 MB
  float* deg    = summed + (long)NN * D;        // N floats

  long zn = (long)NN * D + NN;
  GN_zero_ws<<<2048, 256, 0, stream>>>(summed, zn);

  GN_scatter<<<2048, 256, 0, stream>>>(x, src, dst, summed, deg);

  GN_sage_gemm<<<NN / 16, 256, 0, stream>>>(x, summed, deg,
                                            Wself, Wneigh, bias, out);
}